// MultiGAT_15135464751428
// MI455X (gfx1250) — compile-verified
//
#include <hip/hip_runtime.h>

typedef __attribute__((ext_vector_type(2))) float v2f;
typedef __attribute__((ext_vector_type(8))) float v8f;

namespace {
constexpr int kG   = 7;
constexpr int kN   = 50000;   // nodes (divisible by 16)
constexpr int kE   = 800000;  // edges
constexpr int kF   = 128;     // Fin == H*hid == 128 for both layers
constexpr int kH   = 8;
constexpr int kHid = 16;
}

__device__ __forceinline__ float leakyf(float x) { return x > 0.f ? x : 0.2f * x; }
__device__ __forceinline__ float seluf(float v) {
  const float kScale = 1.0507009873554804934193349852946f;
  const float kAlpha = 1.6732632423543772848170429916717f;
  return v > 0.f ? kScale * v : kScale * kAlpha * (__expf(v) - 1.f);
}
__device__ __forceinline__ bool finitef(float v) {
  return __builtin_fabsf(v) < __builtin_inff();   // false for NaN and +-inf
}
// Monotonic-encoding float atomic max (init must be -inf).
__device__ __forceinline__ void atomic_max_f32(float* addr, float v) {
  if (v >= 0.f) atomicMax((int*)addr, __float_as_int(v));
  else          atomicMin((unsigned int*)addr, __float_as_uint(v));
}

__global__ void fill_kernel(float* p, long n, float v) {
  long i = (long)blockIdx.x * blockDim.x + threadIdx.x;
  long stride = (long)gridDim.x * blockDim.x;
  for (; i < n; i += stride) p[i] = v;
}

// D[N,128] = A[N,128] @ B[128,128] in exact f32 via V_WMMA_F32_16X16X4_F32.
// Each wave owns a 16-row x 128-col strip (8 x v8f accumulators).
// B is staged in LDS *pair-interleaved*: Bs2[q*128 + c] = {B[2q][c], B[2q+1][c]}
// so a lane's 2-dword B fragment is one contiguous 8B ds_load_b64 landing in an
// even-aligned VGPR pair -> no repack movs before v_wmma.
// A fragments are software-pipelined (fetch K+4 before consuming K).
// f32 A 16x4 frag: v0 = A[M=lane%16][k + 2*(lane>=16)], v1 = next K.
// f32 B 4x16 frag (mirrored): v0 = B[k + 2*(lane>=16)][N=lane%16], v1 = next K.
// f32 C/D 16x16: vgpr m = row (m + 8*(lane>=16)), lane%16 = col.
__global__ __launch_bounds__(256) void gemm128_wmma(const float* __restrict__ A,
                                                    const float* __restrict__ B,
                                                    float* __restrict__ D,
                                                    int nrows) {
  __shared__ float2 Bs2[(kF / 2) * kF];      // 65536 bytes, pair-interleaved
  const int tid = threadIdx.x;
  #pragma unroll 4
  for (int i0 = 0; i0 < (kF * kF / 2) / 256; ++i0) {
    int i = tid + i0 * 256;                  // i = q*128 + c
    int q = i >> 7;
    int c = i & 127;
    float2 v;
    v.x = B[(2 * q) * kF + c];
    v.y = B[(2 * q + 1) * kF + c];
    Bs2[i] = v;
  }
  __syncthreads();

  const int wave = tid >> 5;
  const int lane = tid & 31;
  const int rowTile = blockIdx.x * 8 + wave;
  const int nRowTiles = nrows >> 4;
  if (rowTile >= nRowTiles) return;          // wave-uniform: EXEC stays all-ones

  const int laneM = lane & 15;
  const int hi = lane >> 4;                  // lane half selects K pair
  const float* apBase = A + (long)rowTile * 16 * kF + laneM * kF + 2 * hi;

  v8f acc[8] = {};

  v2f afrag;                                  // prologue: K-step 0 fragment
  afrag.x = apBase[0];
  afrag.y = apBase[1];

  for (int k = 0; k < kF; k += 4) {
    v2f anext = afrag;
    if (k + 4 < kF) {                        // pipeline: start next A fetch now
      const float* ap = apBase + k + 4;
      anext.x = ap[0];
      anext.y = ap[1];
    }
    const float2* bq = Bs2 + ((k >> 1) + hi) * kF + laneM;   // this lane's q-row
    #pragma unroll
    for (int t = 0; t < 8; ++t) {
      float2 bv = bq[t * 16];                // single ds_load_b64 per fragment
      v2f bfrag;
      bfrag.x = bv.x;
      bfrag.y = bv.y;
      acc[t] = __builtin_amdgcn_wmma_f32_16x16x4_f32(
          false, afrag, false, bfrag, (short)0, acc[t], false, false);
    }
    afrag = anext;
  }

  #pragma unroll
  for (int t = 0; t < 8; ++t) {
    float* dp = D + ((long)rowTile * 16 + 8 * hi) * kF + t * 16 + laneM;
    #pragma unroll
    for (int m = 0; m < 8; ++m) dp[(long)m * kF] = acc[t][m];
  }
}

// el[n,h] = dot(feat[n,h,:], al[h,:]) ; er likewise. One thread per (n,h).
__global__ void attn_coef_kernel(const float* __restrict__ feat,
                                 const float* __restrict__ al,
                                 const float* __restrict__ ar,
                                 float* __restrict__ el, float* __restrict__ er) {
  int idx = blockIdx.x * blockDim.x + threadIdx.x;
  if (idx >= kN * kH) return;
  int h = idx & (kH - 1);
  int n = idx >> 3;
  const float* f = feat + (long)n * kF + h * kHid;
  const float* a = al + h * kHid;
  const float* r = ar + h * kHid;
  float se = 0.f, sr = 0.f;
  #pragma unroll
  for (int d = 0; d < kHid; ++d) { float v = f[d]; se += v * a[d]; sr += v * r[d]; }
  el[idx] = se;
  er[idx] = sr;
}

__global__ void edge_max_kernel(const int* __restrict__ src, const int* __restrict__ dst,
                                const float* __restrict__ el, const float* __restrict__ er,
                                float* __restrict__ m) {
  int idx = blockIdx.x * blockDim.x + threadIdx.x;   // e*kH + h
  if (idx >= kE * kH) return;
  int h = idx & (kH - 1);
  int e = idx >> 3;
  int s = src[e], d = dst[e];
  float sc = leakyf(el[s * kH + h] + er[d * kH + h]);
  atomic_max_f32(&m[d * kH + h], sc);
}

__global__ void fix_max_kernel(float* __restrict__ m) {
  int idx = blockIdx.x * blockDim.x + threadIdx.x;
  if (idx >= kN * kH) return;
  float v = m[idx];
  if (!finitef(v)) m[idx] = 0.f;               // zero-in-degree nodes
}

__global__ void edge_expsum_kernel(const int* __restrict__ src, const int* __restrict__ dst,
                                   const float* __restrict__ el, const float* __restrict__ er,
                                   const float* __restrict__ m, float* __restrict__ sbuf) {
  int idx = blockIdx.x * blockDim.x + threadIdx.x;
  if (idx >= kE * kH) return;
  int h = idx & (kH - 1);
  int e = idx >> 3;
  int s = src[e], d = dst[e];
  float sc = leakyf(el[s * kH + h] + er[d * kH + h]);
  atomicAdd(&sbuf[d * kH + h], __expf(sc - m[d * kH + h]));
}

// One wave per edge: 32 lanes x float4 = 128 channels; lane's head = lane/4.
// alpha recomputed per lane (cheaper than storing a 25 MB [E,H] array).
// The 512B feat[src] row gather is the dominant traffic: kick it off with a
// global_prefetch before the attention-coefficient gathers and exp/div.
__global__ void edge_agg_kernel(const int* __restrict__ src, const int* __restrict__ dst,
                                const float* __restrict__ el, const float* __restrict__ er,
                                const float* __restrict__ m, const float* __restrict__ sbuf,
                                const float* __restrict__ feat, float* __restrict__ agg) {
  int gid = blockIdx.x * blockDim.x + threadIdx.x;
  int e = gid >> 5;
  if (e >= kE) return;
  int lane = gid & 31;
  int s = src[e], d = dst[e];
  const float* fp = feat + (long)s * kF + lane * 4;
  __builtin_prefetch(fp, 0, 3);                    // global_prefetch_b8
  int h = lane >> 2;
  float sc = leakyf(el[s * kH + h] + er[d * kH + h]);
  float alpha = __expf(sc - m[d * kH + h]) / sbuf[d * kH + h];
  const float4 f = *(const float4*)fp;
  float* o = agg + (long)d * kF + lane * 4;
  atomicAdd(o + 0, alpha * f.x);
  atomicAdd(o + 1, alpha * f.y);
  atomicAdd(o + 2, alpha * f.z);
  atomicAdd(o + 3, alpha * f.w);
}

__global__ void bias_selu_kernel(float* __restrict__ buf, const float* __restrict__ b) {
  long idx = (long)blockIdx.x * blockDim.x + threadIdx.x;
  if (idx >= (long)kN * kF) return;
  int c = (int)(idx & (kF - 1));
  buf[idx] = seluf(buf[idx] + b[c]);
}

// Layer-2 epilogue: selu(agg + b2), mean over 8 heads, * mask[n], block-reduce
// into hs[16] (sum over nodes; /N and /G applied in the final kernel).
__global__ __launch_bounds__(256) void head_mean_reduce_kernel(
    const float* __restrict__ agg, const float* __restrict__ b,
    const float* __restrict__ mask, float* __restrict__ hs) {
  __shared__ float red[256];
  int n = blockIdx.x * 16 + (threadIdx.x >> 4);   // kN % 16 == 0 -> always valid
  int d = threadIdx.x & 15;
  float acc = 0.f;
  #pragma unroll
  for (int h = 0; h < kH; ++h)
    acc += seluf(agg[(long)n * kF + h * kHid + d] + b[h * kHid + d]);
  acc = acc * (1.f / kH) * mask[n];
  red[threadIdx.x] = acc;
  __syncthreads();
  if (threadIdx.x < 16) {
    float t = 0.f;
    #pragma unroll
    for (int j = 0; j < 16; ++j) t += red[threadIdx.x + j * 16];
    atomicAdd(&hs[threadIdx.x], t);
  }
}

__global__ void final_kernel(const float* __restrict__ hs, const float* __restrict__ Wc,
                             const float* __restrict__ bc, float* __restrict__ out) {
  if (threadIdx.x != 0 || blockIdx.x != 0) return;
  float hov[16];
  #pragma unroll
  for (int d = 0; d < 16; ++d) {
    float t = 0.f;
    for (int g = 0; g < kG; ++g) t += hs[g * 16 + d];
    hov[d] = t / ((float)kN * (float)kG);
  }
  float l0 = bc[0], l1 = bc[1];
  #pragma unroll
  for (int d = 0; d < 16; ++d) { l0 += hov[d] * Wc[d * 2 + 0]; l1 += hov[d] * Wc[d * 2 + 1]; }
  float mx = fmaxf(l0, l1);
  float e0 = __expf(l0 - mx), e1 = __expf(l1 - mx);
  float s = e0 + e1;
  out[0] = e0 / s;
  out[1] = e1 / s;
}

extern "C" void kernel_launch(void* const* d_in, const int* in_sizes, int n_in,
                              void* d_out, int out_size, void* d_ws, size_t ws_size,
                              hipStream_t stream) {
  const float* x    = (const float*)d_in[0];
  const int*   esrc = (const int*)d_in[1];
  const int*   edst = (const int*)d_in[2];
  const float* mask = (const float*)d_in[3];
  const float* W1   = (const float*)d_in[4];
  const float* al1  = (const float*)d_in[5];
  const float* ar1  = (const float*)d_in[6];
  const float* b1   = (const float*)d_in[7];
  const float* W2   = (const float*)d_in[8];
  const float* al2  = (const float*)d_in[9];
  const float* ar2  = (const float*)d_in[10];
  const float* b2   = (const float*)d_in[11];
  const float* Wc   = (const float*)d_in[12];
  const float* bc   = (const float*)d_in[13];

  float* ws = (float*)d_ws;
  const long NF = (long)kN * kF;               // 6.4M floats per big buffer
  float* F0 = ws;                              // feat (layer output of GEMM)
  float* F1 = F0 + NF;                         // agg1 -> h1 (layer-2 input)
  float* F2 = F1 + NF;                         // agg2
  float* el = F2 + NF;
  float* er = el + (long)kN * kH;
  float* mb = er + (long)kN * kH;
  float* sb = mb + (long)kN * kH;
  float* hs = sb + (long)kN * kH;              // [7,16] per-graph node sums

  const dim3 blk(256);
  const int gGemm = (kN / 16 + 7) / 8;         // 391 blocks, 8 waves each
  const int gNH   = (kN * kH + 255) / 256;
  const int gEH   = (kE * kH + 255) / 256;
  const int gEW   = (kE * 32 + 255) / 256;     // wave per edge
  const int gNF   = (int)((NF + 255) / 256);
  const int gRed  = kN / 16;
  const float negInf = -__builtin_inff();

  fill_kernel<<<1, blk, 0, stream>>>(hs, kG * 16, 0.f);

  for (int g = 0; g < kG; ++g) {
    const float* xg = x + g * NF;
    const int* sg = esrc + (long)g * kE;
    const int* dg = edst + (long)g * kE;

    // -------- layer 1 --------
    gemm128_wmma<<<gGemm, blk, 0, stream>>>(xg, W1 + g * kF * kF, F0, kN);
    attn_coef_kernel<<<gNH, blk, 0, stream>>>(F0, al1 + g * kF, ar1 + g * kF, el, er);
    fill_kernel<<<256, blk, 0, stream>>>(mb, (long)kN * kH, negInf);
    edge_max_kernel<<<gEH, blk, 0, stream>>>(sg, dg, el, er, mb);
    fix_max_kernel<<<gNH, blk, 0, stream>>>(mb);
    fill_kernel<<<256, blk, 0, stream>>>(sb, (long)kN * kH, 0.f);
    edge_expsum_kernel<<<gEH, blk, 0, stream>>>(sg, dg, el, er, mb, sb);
    fill_kernel<<<256, blk, 0, stream>>>(F1, NF, 0.f);
    edge_agg_kernel<<<gEW, blk, 0, stream>>>(sg, dg, el, er, mb, sb, F0, F1);
    bias_selu_kernel<<<gNF, blk, 0, stream>>>(F1, b1 + g * kF);

    // -------- layer 2 --------
    gemm128_wmma<<<gGemm, blk, 0, stream>>>(F1, W2 + g * kF * kF, F0, kN);
    attn_coef_kernel<<<gNH, blk, 0, stream>>>(F0, al2 + g * kF, ar2 + g * kF, el, er);
    fill_kernel<<<256, blk, 0, stream>>>(mb, (long)kN * kH, negInf);
    edge_max_kernel<<<gEH, blk, 0, stream>>>(sg, dg, el, er, mb);
    fix_max_kernel<<<gNH, blk, 0, stream>>>(mb);
    fill_kernel<<<256, blk, 0, stream>>>(sb, (long)kN * kH, 0.f);
    edge_expsum_kernel<<<gEH, blk, 0, stream>>>(sg, dg, el, er, mb, sb);
    fill_kernel<<<256, blk, 0, stream>>>(F2, NF, 0.f);
    edge_agg_kernel<<<gEW, blk, 0, stream>>>(sg, dg, el, er, mb, sb, F0, F2);
    head_mean_reduce_kernel<<<gRed, blk, 0, stream>>>(F2, b2 + g * kF, mask, hs + g * 16);
  }

  final_kernel<<<1, 32, 0, stream>>>(hs, Wc, bc, (float*)d_out);
}